// NonTrainableNeuralSFA_roadR_29377576305412
// MI455X (gfx1250) — compile-verified
//
#include <hip/hip_runtime.h>
#include <hip/hip_bf16.h>
#include <math.h>

// Problem constants (from reference): B,T,D,V,G,M,N
#define Bn 32
#define Tn 256
#define Dn 2048
#define Vn 41
#define Gn 40
#define Mn 4
#define Nn 10

#define KC 128   // K-chunk staged in LDS
#define NP 48    // padded column count (3 x 16-wide WMMA tiles)

typedef __attribute__((ext_vector_type(2))) float v2f;
typedef __attribute__((ext_vector_type(8))) float v8f;

// ---------------------------------------------------------------------------
// Kernel 1: nn_outputs = sigmoid(x @ W + b)
//   x: (B*T, D) row-major, W: (D, V) row-major, out: (B*T, V)
// Each wave computes a 16-row x 48-col tile with V_WMMA_F32_16X16X4_F32.
// The block (8 waves) cooperatively stages W K-chunks (128 x 48, zero-padded
// past V=41) into LDS so B-fragments are shared across waves: L2 traffic for
// W drops from ~192 MB to ~21 MB; x is streamed from HBM exactly once (67 MB,
// ~2.9 us roofline at 23.3 TB/s).
// ---------------------------------------------------------------------------
__global__ __launch_bounds__(256) void gemm_sigmoid_kernel(
    const float* __restrict__ x, const float* __restrict__ W,
    const float* __restrict__ bias, float* __restrict__ nn_out)
{
    __shared__ float Ws[KC][NP];   // 24 KB

    const int tid  = threadIdx.x;
    const int lane = tid & 31;
    const int wave = tid >> 5;
    const int row0 = blockIdx.x * 128 + wave * 16;

    const int r     = lane & 15;        // row (A) / col (B,C) within tile
    const int khalf = (lane >> 4) * 2;  // K sub-block selected by lane half

    v8f acc[3];
#pragma unroll
    for (int ct = 0; ct < 3; ++ct) acc[ct] = (v8f){};

    const float* xrow = x + (size_t)(row0 + r) * Dn;

    for (int kk0 = 0; kk0 < Dn; kk0 += KC) {
        // ---- stage W[kk0 .. kk0+KC) x 48 into LDS (zero-padded cols) ----
        __syncthreads();   // previous chunk fully consumed
#pragma unroll
        for (int i = tid; i < KC * NP; i += 256) {
            int k = i / NP;
            int n = i - k * NP;
            Ws[k][n] = (n < Vn) ? W[(size_t)(kk0 + k) * Vn + n] : 0.0f;
        }
        __syncthreads();   // chunk visible to all waves

        // ---- compute over the chunk ----
#pragma unroll 4
        for (int k2 = 0; k2 < KC; k2 += 4) {
            // A fragment: A[r][khalf], A[r][khalf+1] -> one aligned 8B load
            v2f a = *(const v2f*)(xrow + kk0 + k2 + khalf);
            const int kb = k2 + khalf;
#pragma unroll
            for (int ct = 0; ct < 3; ++ct) {
                // B fragment from LDS: VGPR j holds W[kb + j][col]
                v2f bf;
                bf.x = Ws[kb][ct * 16 + r];
                bf.y = Ws[kb + 1][ct * 16 + r];
                acc[ct] = __builtin_amdgcn_wmma_f32_16x16x4_f32(
                    false, a, false, bf, (short)0, acc[ct], false, false);
            }
        }
    }

    // Epilogue: C/D layout -> VGPR j = row (j + 8*(lane>>4)), col = lane&15
    const int mhalf = (lane >> 4) * 8;
#pragma unroll
    for (int ct = 0; ct < 3; ++ct) {
        int n = ct * 16 + r;
        if (n < Vn) {
            float bb = bias[n];
#pragma unroll
            for (int j = 0; j < 8; ++j) {
                int m   = row0 + mhalf + j;
                float z = acc[ct][j] + bb;
                // sigmoid via native exp + rcp (v_exp_f32 / v_rcp_f32)
                nn_out[(size_t)m * Vn + n] =
                    __builtin_amdgcn_rcpf(1.0f + __expf(-z));
            }
        }
    }
}

// ---------------------------------------------------------------------------
// Kernel 2: guard_probs[bt, g] = sum_m coef[g,m] * prod_v fma(pos-neg, p, 1-pos)
// One thread per (bt, g); grid = (BT/64, G). Mask/coef addresses are uniform
// across the wave (scalarizable); nn_outputs is L2-resident (1.3 MB).
// ---------------------------------------------------------------------------
__global__ __launch_bounds__(64) void guard_kernel(
    const float* __restrict__ nn, const float* __restrict__ pos,
    const float* __restrict__ neg, const float* __restrict__ coef,
    float* __restrict__ guard)
{
    const int bt = blockIdx.x * 64 + threadIdx.x;
    const int g  = blockIdx.y;
    const float* p = nn + (size_t)bt * Vn;

    float acc = 0.0f;
#pragma unroll
    for (int m = 0; m < Mn; ++m) {
        const float* pm = pos + ((size_t)g * Mn + m) * Vn;
        const float* nm = neg + ((size_t)g * Mn + m) * Vn;
        float mono = 1.0f;
#pragma unroll
        for (int v = 0; v < Vn; ++v) {
            float po = pm[v];
            float ne = nm[v];
            // pos*p + neg*(1-p) + (1-pos-neg) == (pos-neg)*p + (1-pos)
            mono *= fmaf(po - ne, p[v], 1.0f - po);
        }
        acc = fmaf(coef[g * Mn + m], mono, acc);
    }
    guard[(size_t)bt * Gn + g] = acc;
}

// ---------------------------------------------------------------------------
// Kernel 3: sequential state scan per batch. One wave (32 threads) per b.
// T is sparse: 40 (from,to) guard entries + forced T[9][9]=1 (which overrides
// any (9,9) guard pair, matching the reference's .set order). Barriers in a
// single-wave workgroup lower to S_NOP; only LDS waits remain.
// ---------------------------------------------------------------------------
__global__ __launch_bounds__(32) void scan_kernel(
    const float* __restrict__ guard, const int* __restrict__ from_idx,
    const int* __restrict__ to_idx, float* __restrict__ out_state)
{
    const int b    = blockIdx.x;
    const int lane = threadIdx.x;

    __shared__ float s[16];
    __shared__ float snew[16];
    __shared__ int   fidx[Gn];
    __shared__ int   tidx[Gn];

    if (lane < 16) s[lane] = (lane == 0) ? 1.0f : 0.0f;
    for (int g = lane; g < Gn; g += 32) {
        fidx[g] = from_idx[g];
        tidx[g] = to_idx[g];
    }
    __syncthreads();

    for (int tt = 0; tt < Tn; ++tt) {
        if (lane < 16) snew[lane] = 0.0f;
        __syncthreads();

        const float* gp = guard + ((size_t)b * Tn + tt) * Gn;
        for (int g = lane; g < Gn; g += 32) {
            int f = fidx[g];
            int t = tidx[g];
            if (!(f == (Nn - 1) && t == (Nn - 1))) {   // (9,9) overridden to 1
                atomicAdd(&snew[t], s[f] * gp[g]);
            }
        }
        if (lane == 0) atomicAdd(&snew[Nn - 1], s[Nn - 1]);  // T[9][9] = 1
        __syncthreads();

        if (lane < 16) s[lane] = snew[lane];
        __syncthreads();
    }

    if (lane < Nn) out_state[(size_t)b * Nn + lane] = s[lane];
}

// ---------------------------------------------------------------------------
extern "C" void kernel_launch(void* const* d_in, const int* in_sizes, int n_in,
                              void* d_out, int out_size, void* d_ws, size_t ws_size,
                              hipStream_t stream) {
    const float* seq  = (const float*)d_in[0];  // (B,1,T,D)
    // d_in[1] = bboxes — unused by the reference
    const float* W    = (const float*)d_in[2];  // (D,V)
    const float* bias = (const float*)d_in[3];  // (V,)
    const float* pos  = (const float*)d_in[4];  // (G,M,V)
    const float* neg  = (const float*)d_in[5];  // (G,M,V)
    const float* coef = (const float*)d_in[6];  // (G,M)
    const int*   fi   = (const int*)d_in[7];    // (G,)
    const int*   ti   = (const int*)d_in[8];    // (G,)

    float* nn_out = (float*)d_out;                       // (B,T,V)
    float* guard  = nn_out + (size_t)Bn * Tn * Vn;       // (B,T,G)
    float* state  = guard + (size_t)Bn * Tn * Gn;        // (B,N)

    gemm_sigmoid_kernel<<<dim3((Bn * Tn) / 128), dim3(256), 0, stream>>>(
        seq, W, bias, nn_out);

    guard_kernel<<<dim3((Bn * Tn) / 64, Gn), dim3(64), 0, stream>>>(
        nn_out, pos, neg, coef, guard);

    scan_kernel<<<dim3(Bn), dim3(32), 0, stream>>>(guard, fi, ti, state);
}